// BoxLoss_33724083208421
// MI455X (gfx1250) — compile-verified
//
#include <hip/hip_runtime.h>
#include <math.h>

#define BB 64
#define PP 8732
#define OO 32
#define CC 81
#define EPSF 1e-6f
#define THRESH 0.5f
#define NPART 4096   // 64 blocks/batch * 64 batches

typedef __attribute__((ext_vector_type(2))) float v2f;
typedef __attribute__((ext_vector_type(8))) float v8f;

__device__ __forceinline__ unsigned long long packIoU(float iou, int p) {
    // iou >= 0 so float bits are monotonic; ~p makes ties prefer smallest p
    unsigned int fb = __float_as_uint(iou);
    return ((unsigned long long)fb << 32) |
           (unsigned long long)(0xFFFFFFFFu - (unsigned)p);
}

// ---------------- Kernel 0: zero workspace ----------------
__global__ void k_init(unsigned long long* __restrict__ bp,
                       float* __restrict__ partials) {
    int i = blockIdx.x * blockDim.x + threadIdx.x;
    if (i < BB * OO) bp[i] = 0ull;
    if (i < 4 * NPART) partials[i] = 0.0f;
}

// ---------------- Kernel 1: best prior per (batch, object) ----------------
// lane l owns object l (wave32 == 32 objects); waves stream priors.
__global__ void k_bestprior(const float* __restrict__ priors,   // [P,4] cxcywh
                            const float* __restrict__ tboxes,   // [B,O,4] xywh
                            unsigned long long* __restrict__ bp) {
    const int b    = blockIdx.y;
    const int lane = threadIdx.x & 31;
    const int wave = blockIdx.x * (blockDim.x >> 5) + (threadIdx.x >> 5);
    const int nwav = gridDim.x * (blockDim.x >> 5);

    float4 t = ((const float4*)tboxes)[b * OO + lane];     // xywh
    const float tx0 = t.x, ty0 = t.y, tx1 = t.x + t.z, ty1 = t.y + t.w;
    const float tarea = (tx1 - tx0) * (ty1 - ty0);

    float bestI = -1.0f;
    int   bestP = 0;
    for (int p = wave; p < PP; p += nwav) {
        float4 pr = ((const float4*)priors)[p];            // broadcast load
        float ax0 = pr.x - pr.z * 0.5f, ay0 = pr.y - pr.w * 0.5f;
        float ax1 = pr.x + pr.z * 0.5f, ay1 = pr.y + pr.w * 0.5f;
        float w = fmaxf(fminf(ax1, tx1) - fmaxf(ax0, tx0), 0.0f);
        float h = fmaxf(fminf(ay1, ty1) - fmaxf(ay0, ty0), 0.0f);
        float inter = w * h;
        float aarea = (ax1 - ax0) * (ay1 - ay0);
        float iou = inter / (aarea + tarea - inter + EPSF);
        if (iou > bestI) { bestI = iou; bestP = p; }       // strict > keeps smallest p
    }
    atomicMax(&bp[b * OO + lane], packIoU(bestI, bestP));
}

// ---------------- Kernel 2: fused matching + CE + L1 + IoU loss ----------------
__global__ void k_loss(const float* __restrict__ plocs,   // [B,P,4] cxcywh
                       const float* __restrict__ pcls,    // [B,P,C]
                       const float* __restrict__ priors,  // [P,4]  cxcywh
                       const float* __restrict__ tboxes,  // [B,O,4] xywh
                       const int*   __restrict__ tlabels, // [B,O]
                       const unsigned long long* __restrict__ bp,
                       float* __restrict__ partials) {
    __shared__ float4  sT[OO];      // target xyxy
    __shared__ int     sL[OO];
    __shared__ unsigned sBP[OO];    // force-matched prior per object
    __shared__ float   sAcc[8][4];

    const int b    = blockIdx.y;
    const int lane = threadIdx.x & 31;
    const int wid  = threadIdx.x >> 5;

    if (threadIdx.x < OO) {
        float4 t = ((const float4*)tboxes)[b * OO + threadIdx.x];
        sT[threadIdx.x] = make_float4(t.x, t.y, t.x + t.z, t.y + t.w);
        sL[threadIdx.x] = tlabels[b * OO + threadIdx.x];
        sBP[threadIdx.x] = 0xFFFFFFFFu -
            (unsigned)(bp[b * OO + threadIdx.x] & 0xFFFFFFFFull);
    }
    __syncthreads();

    const int chunk = (PP + gridDim.x - 1) / gridDim.x;   // 137
    const int pBeg = blockIdx.x * chunk;
    const int pEnd = min(pBeg + chunk, PP);

    const float4   T    = sT[lane];
    const int      TL   = sL[lane];
    const unsigned TBP  = sBP[lane];
    const float    tarea = (T.z - T.x) * (T.w - T.y);

    float accCE = 0.f, accL1 = 0.f, accIL = 0.f, accN = 0.f;

    for (int p = pBeg + wid; p < pEnd; p += 8) {
        // ---- matching: lane == object ----
        float4 pr = ((const float4*)priors)[p];
        float ax0 = pr.x - pr.z * 0.5f, ay0 = pr.y - pr.w * 0.5f;
        float ax1 = pr.x + pr.z * 0.5f, ay1 = pr.y + pr.w * 0.5f;
        float w = fmaxf(fminf(ax1, T.z) - fmaxf(ax0, T.x), 0.0f);
        float h = fmaxf(fminf(ay1, T.w) - fmaxf(ay0, T.y), 0.0f);
        float inter = w * h;
        float aarea = (ax1 - ax0) * (ay1 - ay0);
        float iou = inter / (aarea + tarea - inter + EPSF);

        float bI = iou; int bO = lane;                     // argmax over objects
        for (int off = 16; off; off >>= 1) {
            float oi = __shfl_xor(bI, off, 32);
            int   oo = __shfl_xor(bO, off, 32);
            if (oi > bI || (oi == bI && oo < bO)) { bI = oi; bO = oo; }
        }
        unsigned long long fm = __ballot(TBP == (unsigned)p);
        if (fm) { bO = 63 - __clzll(fm); bI = 1.0f; }      // last-write-wins scatter

        int label = (bI < THRESH) ? 0 : __shfl(TL, bO, 32);
        float4 tb = sT[bO];
        bool pos = (label != 0);

        // ---- cross entropy: wave-cooperative log-softmax over 81 classes ----
        const float* row = pcls + ((size_t)b * PP + p) * CC;
        float l0 = row[lane];
        float l1 = row[lane + 32];
        float l2 = (lane < CC - 64) ? row[lane + 64] : -__builtin_inff();
        float m = fmaxf(l0, fmaxf(l1, l2));
        for (int off = 16; off; off >>= 1) m = fmaxf(m, __shfl_xor(m, off, 32));
        float s = __expf(l0 - m) + __expf(l1 - m) + __expf(l2 - m);
        for (int off = 16; off; off >>= 1) s += __shfl_xor(s, off, 32);
        int slot = label >> 5, srcl = label & 31;
        float v0 = __shfl(l0, srcl, 32);
        float v1 = __shfl(l1, srcl, 32);
        float v2 = __shfl(l2, srcl, 32);
        float ll = (slot == 0) ? v0 : ((slot == 1) ? v1 : v2);
        accCE += (m + logf(s)) - ll;

        if (pos) {   // pos is wave-uniform
            accN += 1.0f;
            float4 pl = ((const float4*)plocs)[(size_t)b * PP + p];
            float tcx = 0.5f * (tb.x + tb.z), tcy = 0.5f * (tb.y + tb.w);
            float tw = tb.z - tb.x, th = tb.w - tb.y;
            accL1 += fabsf(pl.x - tcx) + fabsf(pl.y - tcy) +
                     fabsf(pl.z - tw)  + fabsf(pl.w - th);
            float px0 = pl.x - pl.z * 0.5f, py0 = pl.y - pl.w * 0.5f;
            float px1 = pl.x + pl.z * 0.5f, py1 = pl.y + pl.w * 0.5f;
            float iw = fmaxf(fminf(px1, tb.z) - fmaxf(px0, tb.x), 0.0f);
            float ih = fmaxf(fminf(py1, tb.w) - fmaxf(py0, tb.y), 0.0f);
            float ii = iw * ih;
            float pa = (px1 - px0) * (py1 - py0);
            accIL += 1.0f - ii / (tw * th + pa - ii + EPSF);
        }
    }

    if (lane == 0) {
        sAcc[wid][0] = accCE; sAcc[wid][1] = accL1;
        sAcc[wid][2] = accIL; sAcc[wid][3] = accN;
    }
    __syncthreads();
    if (threadIdx.x == 0) {
        float t0 = 0.f, t1 = 0.f, t2 = 0.f, t3 = 0.f;
        for (int wv = 0; wv < 8; ++wv) {
            t0 += sAcc[wv][0]; t1 += sAcc[wv][1];
            t2 += sAcc[wv][2]; t3 += sAcc[wv][3];
        }
        int bid = blockIdx.y * gridDim.x + blockIdx.x;
        partials[0 * NPART + bid] = t0;   // CE
        partials[1 * NPART + bid] = t1;   // L1
        partials[2 * NPART + bid] = t2;   // IoU loss
        partials[3 * NPART + bid] = t3;   // n_pos
    }
}

// ---------------- Kernel 3: WMMA tree-reduction + finalize ----------------
// Sum 4096 partials per quantity via D = A(16x4) * Ones(4x16) + C accumulate:
// every column of D holds the running total -> one shfl_xor(16) finishes it.
__global__ void k_finalize(const float* __restrict__ partials,
                           float* __restrict__ out) {
    const int lane = threadIdx.x;                 // 32 threads, EXEC all ones
    v2f ones; ones[0] = 1.0f; ones[1] = 1.0f;
    // f32 16x4 A layout: lanes 0-15 hold K=0,1 ; lanes 16-31 hold K=2,3
    const int idx0 = (lane & 15) + ((lane >> 4) << 5);
    float tot[4];
    for (int q = 0; q < 4; ++q) {
        v8f c = {0.f, 0.f, 0.f, 0.f, 0.f, 0.f, 0.f, 0.f};
        const float* base = partials + q * NPART;
        for (int ch = 0; ch < NPART / 64; ++ch) {
            v2f a;
            a[0] = base[ch * 64 + idx0];
            a[1] = base[ch * 64 + idx0 + 16];
            c = __builtin_amdgcn_wmma_f32_16x16x4_f32(
                    false, a, false, ones, (short)0, c, false, false);
        }
        float partl = c[0] + c[1] + c[2] + c[3] + c[4] + c[5] + c[6] + c[7];
        partl += __shfl_xor(partl, 16, 32);
        tot[q] = partl;
    }
    if (lane == 0) {
        float ce = tot[0], l1 = tot[1], il = tot[2], n = tot[3];
        out[0] = l1 / (n * 4.0f);                 // loc_loss
        out[1] = ce / (float)(BB * PP);           // cross_loss
        out[2] = il / n;                          // iou_loss
    }
}

extern "C" void kernel_launch(void* const* d_in, const int* in_sizes, int n_in,
                              void* d_out, int out_size, void* d_ws, size_t ws_size,
                              hipStream_t stream) {
    (void)in_sizes; (void)n_in; (void)out_size; (void)ws_size;
    const float* plocs  = (const float*)d_in[0];   // [B,P,4]
    const float* pcls   = (const float*)d_in[1];   // [B,P,C]
    const float* priors = (const float*)d_in[2];   // [P,4]
    const float* tboxes = (const float*)d_in[3];   // [B,O,4]
    const int*   tlabs  = (const int*)d_in[4];     // [B,O]

    unsigned long long* bp = (unsigned long long*)d_ws;                 // 16 KB
    float* partials = (float*)((char*)d_ws + BB * OO * sizeof(unsigned long long)); // 64 KB
    float* out = (float*)d_out;

    k_init<<<64, 256, 0, stream>>>(bp, partials);

    dim3 g1(16, BB);
    k_bestprior<<<g1, 256, 0, stream>>>(priors, tboxes, bp);

    dim3 g2(64, BB);
    k_loss<<<g2, 256, 0, stream>>>(plocs, pcls, priors, tboxes, tlabs, bp, partials);

    k_finalize<<<1, 32, 0, stream>>>(partials, out);
}